// Loss_47373489275493
// MI455X (gfx1250) — compile-verified
//
#include <hip/hip_runtime.h>
#include <hip/hip_bf16.h>
#include <math.h>

// Problem constants (match reference)
#define B_ROWS 2048
#define N_ROWS 4096          // 2*B
#define D_DIM  1024
#define G2I_INV_TEMP 20.0f   // 1 / 0.05

// Column chunking for occupancy: partial (m,l) per (row, chunk), merged later.
#define CHUNK_COLS   512
#define L2I_CHUNKS   (B_ROWS / CHUNK_COLS)   // 4
#define G2I_CHUNKS   (N_ROWS / CHUNK_COLS)   // 8

typedef __attribute__((ext_vector_type(2))) float v2f;
typedef __attribute__((ext_vector_type(8))) float v8f;

// ---------------------------------------------------------------------------
// WMMA fragment load for S = X @ Y^T with row-major X [M,K], Y [N,K].
// V_WMMA_F32_16X16X4_F32: A 16x4 (lane->M, {vgpr,half}->K), B 4x16 (lane->N).
// Both operands index identically into row-major storage:
//   lane L reads float2 at [row0 + (L&15)][k0 + 2*(L>>4)]
// ---------------------------------------------------------------------------
__device__ __forceinline__ v2f load_frag(const float* __restrict__ base,
                                         int row0, int k0, int lane, int ldk) {
  const int half = lane >> 4;
  const int lr   = lane & 15;
  return *reinterpret_cast<const v2f*>(base + (size_t)(row0 + lr) * ldk +
                                       (k0 + 2 * half));
}

// Reductions across one 16-lane half of the wave (masks < 16 keep halves
// separate under wave32), matching the C layout: VGPR r holds row r for lanes
// 0-15 and row r+8 for lanes 16-31; column = lane & 15.
__device__ __forceinline__ float half16_max(float v) {
#pragma unroll
  for (int off = 1; off < 16; off <<= 1) v = fmaxf(v, __shfl_xor(v, off, 32));
  return v;
}
__device__ __forceinline__ float half16_sum(float v) {
#pragma unroll
  for (int off = 1; off < 16; off <<= 1) v += __shfl_xor(v, off, 32);
  return v;
}
__device__ __forceinline__ float wave_sum(float v) {
#pragma unroll
  for (int off = 1; off < 32; off <<= 1) v += __shfl_xor(v, off, 32);
  return v;
}

// Double-buffered 16x64 GEMM micro-tile over the full K, accumulating into
// c0..c3.  10 loads (two k-steps) are issued ahead of each 8-WMMA burst so
// L2 latency overlaps the matrix pipe instead of a full wait per 4 WMMAs.
__device__ __forceinline__ void gemm_strip_64(const float* __restrict__ Arow,
                                              const float* __restrict__ Bcol,
                                              int lane, int K,
                                              v8f& c0, v8f& c1, v8f& c2, v8f& c3) {
  for (int k0 = 0; k0 < K; k0 += 8) {
    v2f a0 = load_frag(Arow, 0,  k0,     lane, K);
    v2f p0 = load_frag(Bcol, 0,  k0,     lane, K);
    v2f p1 = load_frag(Bcol, 16, k0,     lane, K);
    v2f p2 = load_frag(Bcol, 32, k0,     lane, K);
    v2f p3 = load_frag(Bcol, 48, k0,     lane, K);
    v2f a1 = load_frag(Arow, 0,  k0 + 4, lane, K);
    v2f q0 = load_frag(Bcol, 0,  k0 + 4, lane, K);
    v2f q1 = load_frag(Bcol, 16, k0 + 4, lane, K);
    v2f q2 = load_frag(Bcol, 32, k0 + 4, lane, K);
    v2f q3 = load_frag(Bcol, 48, k0 + 4, lane, K);
    c0 = __builtin_amdgcn_wmma_f32_16x16x4_f32(false, a0, false, p0, (short)0, c0, false, false);
    c1 = __builtin_amdgcn_wmma_f32_16x16x4_f32(false, a0, false, p1, (short)0, c1, false, false);
    c2 = __builtin_amdgcn_wmma_f32_16x16x4_f32(false, a0, false, p2, (short)0, c2, false, false);
    c3 = __builtin_amdgcn_wmma_f32_16x16x4_f32(false, a0, false, p3, (short)0, c3, false, false);
    c0 = __builtin_amdgcn_wmma_f32_16x16x4_f32(false, a1, false, q0, (short)0, c0, false, false);
    c1 = __builtin_amdgcn_wmma_f32_16x16x4_f32(false, a1, false, q1, (short)0, c1, false, false);
    c2 = __builtin_amdgcn_wmma_f32_16x16x4_f32(false, a1, false, q2, (short)0, c2, false, false);
    c3 = __builtin_amdgcn_wmma_f32_16x16x4_f32(false, a1, false, q3, (short)0, c3, false, false);
  }
}

// ---------------------------------------------------------------------------
// Partial row-LSE of scale * (A @ B^T) over one column chunk.
// grid = (M/128, nChunks); one wave per 16-row strip.  Online (streaming)
// log-sum-exp per row; partial (m, l) written per (row, chunk).
// ---------------------------------------------------------------------------
__global__ void __launch_bounds__(256)
lse_part_plain(const float* __restrict__ A, const float* __restrict__ Bm,
               const float* __restrict__ scale_ptr, int K,
               int colsPerChunk, int nChunks,
               float* __restrict__ pm, float* __restrict__ pl) {
  const int lane  = threadIdx.x & 31;
  const int wave  = blockIdx.x * (blockDim.x >> 5) + (threadIdx.x >> 5);
  const int row0  = wave * 16;
  const int chunk = blockIdx.y;
  const float scale = scale_ptr[0];
  const int half = lane >> 4;

  float m8[8], l8[8];
#pragma unroll
  for (int r = 0; r < 8; ++r) { m8[r] = -INFINITY; l8[r] = 0.0f; }

  const int jend = (chunk + 1) * colsPerChunk;
  for (int j0 = chunk * colsPerChunk; j0 < jend; j0 += 64) {
    v8f c0 = {}, c1 = {}, c2 = {}, c3 = {};
    gemm_strip_64(A + (size_t)row0 * K, Bm + (size_t)j0 * K, lane, K,
                  c0, c1, c2, c3);
    v8f ct[4] = {c0, c1, c2, c3};
#pragma unroll
    for (int t = 0; t < 4; ++t) {
#pragma unroll
      for (int r = 0; r < 8; ++r) {
        float x  = scale * ct[t][r];
        float tm = half16_max(x);
        float mn = fmaxf(m8[r], tm);
        float e  = half16_sum(__expf(x - mn));
        l8[r] = l8[r] * __expf(m8[r] - mn) + e;
        m8[r] = mn;
      }
    }
  }
  if ((lane & 15) == 0) {
#pragma unroll
    for (int r = 0; r < 8; ++r) {
      const int row = row0 + half * 8 + r;
      pm[row * nChunks + chunk] = m8[r];
      pl[row * nChunks + chunk] = l8[r];
    }
  }
}

// ---------------------------------------------------------------------------
// g2i partial row-LSE: rows/cols are concat(gli, img); raw dots scaled by
// rn[row]*rn[col]*invTemp; main diagonal masked to -inf.  Strips (16), column
// groups (64) and chunks (512) never straddle the 2048 boundary.
// ---------------------------------------------------------------------------
__global__ void __launch_bounds__(256)
lse_part_g2i(const float* __restrict__ gli, const float* __restrict__ img,
             const float* __restrict__ rn, float invTemp,
             int colsPerChunk, int nChunks,
             float* __restrict__ pm, float* __restrict__ pl) {
  const int lane  = threadIdx.x & 31;
  const int wave  = blockIdx.x * (blockDim.x >> 5) + (threadIdx.x >> 5);
  const int row0  = wave * 16;
  const int chunk = blockIdx.y;
  const int half  = lane >> 4;
  const int lr    = lane & 15;

  const float* rowbase = (row0 < B_ROWS) ? gli + (size_t)row0 * D_DIM
                                         : img + (size_t)(row0 - B_ROWS) * D_DIM;
  float rr[8], m8[8], l8[8];
#pragma unroll
  for (int r = 0; r < 8; ++r) {
    rr[r] = rn[row0 + half * 8 + r];
    m8[r] = -INFINITY;
    l8[r] = 0.0f;
  }

  const int jend = (chunk + 1) * colsPerChunk;
  for (int j0 = chunk * colsPerChunk; j0 < jend; j0 += 64) {
    const float* colbase = (j0 < B_ROWS) ? gli + (size_t)j0 * D_DIM
                                         : img + (size_t)(j0 - B_ROWS) * D_DIM;
    v8f c0 = {}, c1 = {}, c2 = {}, c3 = {};
    gemm_strip_64(rowbase, colbase, lane, D_DIM, c0, c1, c2, c3);
    v8f ct[4] = {c0, c1, c2, c3};
#pragma unroll
    for (int t = 0; t < 4; ++t) {
      const int col = j0 + 16 * t + lr;
      const float rc = rn[col] * invTemp;
#pragma unroll
      for (int r = 0; r < 8; ++r) {
        const int grow = row0 + half * 8 + r;
        float x = ct[t][r] * rr[r] * rc;
        if (grow == col) x = -INFINITY;          // mask main diagonal
        float tm = half16_max(x);
        float mn = fmaxf(m8[r], tm);
        float e  = half16_sum(__expf(x - mn));
        l8[r] = l8[r] * __expf(m8[r] - mn) + e;
        m8[r] = mn;
      }
    }
  }
  if (lr == 0) {
#pragma unroll
    for (int r = 0; r < 8; ++r) {
      const int row = row0 + half * 8 + r;
      pm[row * nChunks + chunk] = m8[r];
      pl[row * nChunks + chunk] = l8[r];
    }
  }
}

// Merge per-chunk partial (m, l) into final lse per row.
__global__ void __launch_bounds__(256)
merge_lse_kernel(const float* __restrict__ pm, const float* __restrict__ pl,
                 int nChunks, int rows, float* __restrict__ lse_out) {
  const int row = blockIdx.x * blockDim.x + threadIdx.x;
  if (row >= rows) return;
  float M = -INFINITY;
  for (int c = 0; c < nChunks; ++c) M = fmaxf(M, pm[row * nChunks + c]);
  float s = 0.0f;
  for (int c = 0; c < nChunks; ++c)
    s += pl[row * nChunks + c] * __expf(pm[row * nChunks + c] - M);
  lse_out[row] = M + __logf(s);
}

// ---------------------------------------------------------------------------
// 1/||z_row|| for z = concat(gli, img).  One block per row.
// ---------------------------------------------------------------------------
__global__ void __launch_bounds__(256)
row_rnorm_kernel(const float* __restrict__ gli, const float* __restrict__ img,
                 float* __restrict__ rn) {
  const int row = blockIdx.x;
  const float* p = (row < B_ROWS) ? gli + (size_t)row * D_DIM
                                  : img + (size_t)(row - B_ROWS) * D_DIM;
  float s = 0.0f;
  for (int i = threadIdx.x; i < D_DIM; i += blockDim.x) {
    float v = p[i];
    s += v * v;
  }
  __shared__ float red[256];
  red[threadIdx.x] = s;
  __syncthreads();
  for (int st = 128; st > 0; st >>= 1) {
    if (threadIdx.x < st) red[threadIdx.x] += red[threadIdx.x + st];
    __syncthreads();
  }
  if (threadIdx.x == 0) rn[row] = rsqrtf(red[0]);
}

// pos_l2i[i] = logit_scale * dot(img_i, txt_i).  One wave per row.
__global__ void __launch_bounds__(256)
pos_l2i_kernel(const float* __restrict__ img, const float* __restrict__ txt,
               const float* __restrict__ scale_ptr, float* __restrict__ pos) {
  const int lane = threadIdx.x & 31;
  const int row  = blockIdx.x * (blockDim.x >> 5) + (threadIdx.x >> 5);
  if (row >= B_ROWS) return;
  const float* a = img + (size_t)row * D_DIM;
  const float* b = txt + (size_t)row * D_DIM;
  float s = 0.0f;
  for (int i = lane; i < D_DIM; i += 32) s += a[i] * b[i];
  s = wave_sum(s);
  if (lane == 0) pos[row] = scale_ptr[0] * s;
}

// pos_g[i] = dot(z_i, z_{(i+B)%N}) * rn[i]*rn[(i+B)%N] * invTemp.
// The pair is always (gli_r, img_r) with r = i % B.  One wave per row.
__global__ void __launch_bounds__(256)
pos_g2i_kernel(const float* __restrict__ gli, const float* __restrict__ img,
               const float* __restrict__ rn, float invTemp,
               float* __restrict__ pos) {
  const int lane = threadIdx.x & 31;
  const int row  = blockIdx.x * (blockDim.x >> 5) + (threadIdx.x >> 5);
  if (row >= N_ROWS) return;
  const int r = (row < B_ROWS) ? row : row - B_ROWS;
  const int j = (row + B_ROWS) % N_ROWS;
  const float* a = gli + (size_t)r * D_DIM;
  const float* b = img + (size_t)r * D_DIM;
  float s = 0.0f;
  for (int i = lane; i < D_DIM; i += 32) s += a[i] * b[i];
  s = wave_sum(s);
  if (lane == 0) pos[row] = s * rn[row] * rn[j] * invTemp;
}

// Final scalar reduction: out = {total, l2i, g2i}.
__global__ void __launch_bounds__(256)
finalize_kernel(const float* __restrict__ lse_img, const float* __restrict__ lse_txt,
                const float* __restrict__ pos_l2i, const float* __restrict__ lse_g,
                const float* __restrict__ pos_g, float* __restrict__ out) {
  const int tid = threadIdx.x;
  float s12 = 0.0f, s3 = 0.0f, s4 = 0.0f;
  for (int i = tid; i < B_ROWS; i += blockDim.x) {
    s12 += lse_img[i] + lse_txt[i];
    s3  += pos_l2i[i];
  }
  for (int i = tid; i < N_ROWS; i += blockDim.x) s4 += lse_g[i] - pos_g[i];
  __shared__ float r0[256], r1[256], r2[256];
  r0[tid] = s12; r1[tid] = s3; r2[tid] = s4;
  __syncthreads();
  for (int st = 128; st > 0; st >>= 1) {
    if (tid < st) { r0[tid] += r0[tid + st]; r1[tid] += r1[tid + st]; r2[tid] += r2[tid + st]; }
    __syncthreads();
  }
  if (tid == 0) {
    float l2i = 0.5f * (r0[0] - 2.0f * r1[0]) / (float)B_ROWS;
    float g2i = r2[0] / (float)N_ROWS;
    out[0] = l2i + g2i;   // total
    out[1] = l2i;
    out[2] = g2i;
  }
}

// ---------------------------------------------------------------------------
extern "C" void kernel_launch(void* const* d_in, const int* in_sizes, int n_in,
                              void* d_out, int out_size, void* d_ws, size_t ws_size,
                              hipStream_t stream) {
  const float* img = (const float*)d_in[0];
  const float* gli = (const float*)d_in[1];
  const float* txt = (const float*)d_in[2];
  const float* ls  = (const float*)d_in[3];   // device scalar logit_scale
  float* out = (float*)d_out;

  // Workspace layout (floats): ~456 KB total.
  float* ws = (float*)d_ws;
  size_t o = 0;
  float* lse_img = ws + o; o += B_ROWS;
  float* lse_txt = ws + o; o += B_ROWS;
  float* pos_l2i = ws + o; o += B_ROWS;
  float* rn      = ws + o; o += N_ROWS;
  float* lse_g   = ws + o; o += N_ROWS;
  float* pos_g   = ws + o; o += N_ROWS;
  float* pm_img  = ws + o; o += B_ROWS * L2I_CHUNKS;
  float* pl_img  = ws + o; o += B_ROWS * L2I_CHUNKS;
  float* pm_txt  = ws + o; o += B_ROWS * L2I_CHUNKS;
  float* pl_txt  = ws + o; o += B_ROWS * L2I_CHUNKS;
  float* pm_g    = ws + o; o += N_ROWS * G2I_CHUNKS;
  float* pl_g    = ws + o; o += N_ROWS * G2I_CHUNKS;

  // 1) row inverse norms of concat(gli, img)
  row_rnorm_kernel<<<N_ROWS, 256, 0, stream>>>(gli, img, rn);

  // 2) l2i: partial row LSE of s*(img @ txt^T) and s*(txt @ img^T) (== col LSE)
  {
    dim3 grid(B_ROWS / 128, L2I_CHUNKS);
    lse_part_plain<<<grid, 256, 0, stream>>>(img, txt, ls, D_DIM,
                                             CHUNK_COLS, L2I_CHUNKS, pm_img, pl_img);
    lse_part_plain<<<grid, 256, 0, stream>>>(txt, img, ls, D_DIM,
                                             CHUNK_COLS, L2I_CHUNKS, pm_txt, pl_txt);
  }

  // 3) g2i: partial masked row LSE of normalized similarity / temp
  {
    dim3 grid(N_ROWS / 128, G2I_CHUNKS);
    lse_part_g2i<<<grid, 256, 0, stream>>>(gli, img, rn, G2I_INV_TEMP,
                                           CHUNK_COLS, G2I_CHUNKS, pm_g, pl_g);
  }

  // 4) merge chunk partials into final row LSEs
  merge_lse_kernel<<<B_ROWS / 256, 256, 0, stream>>>(pm_img, pl_img, L2I_CHUNKS,
                                                     B_ROWS, lse_img);
  merge_lse_kernel<<<B_ROWS / 256, 256, 0, stream>>>(pm_txt, pl_txt, L2I_CHUNKS,
                                                     B_ROWS, lse_txt);
  merge_lse_kernel<<<N_ROWS / 256, 256, 0, stream>>>(pm_g, pl_g, G2I_CHUNKS,
                                                     N_ROWS, lse_g);

  // 5) positives
  pos_l2i_kernel<<<B_ROWS / 8, 256, 0, stream>>>(img, txt, ls, pos_l2i);
  pos_g2i_kernel<<<N_ROWS / 8, 256, 0, stream>>>(gli, img, rn, G2I_INV_TEMP, pos_g);

  // 6) combine
  finalize_kernel<<<1, 256, 0, stream>>>(lse_img, lse_txt, pos_l2i, lse_g, pos_g, out);
}